// UpsamplerLocalAttention_41918880808979
// MI455X (gfx1250) — compile-verified
//
#include <hip/hip_runtime.h>

// MI455X / gfx1250, wave32. fp32 WMMA (16x16x4) for all GEMM-shaped work,
// async-to-LDS staging + software prefetch on the bandwidth-dominant stream.
typedef __attribute__((ext_vector_type(2))) float v2f;
typedef __attribute__((ext_vector_type(8))) float v8f;

#define CCH 256      // Cc
#define CFH 512      // Cf
#define NCTX 4       // N contexts
#define HSM 64       // h = w
#define HWS 4096     // h*w
#define HWB 16384    // H*W
#define TAPSZ 131072 // CCH*CFH (one conv tap plane after repack)

// ---- CDNA5 async copy helpers (ISA 10.?/15.18: GLOBAL_LOAD_ASYNC_TO_LDS) ----
__device__ __forceinline__ void async_b128_to_lds(unsigned lds_byte_off,
                                                  const float* gptr) {
  asm volatile("global_load_async_to_lds_b128 %0, %1, off"
               :: "v"(lds_byte_off), "v"(gptr) : "memory");
}
__device__ __forceinline__ void wait_asynccnt0() {
  asm volatile("s_wait_asynccnt 0" ::: "memory");
}

// ---------------------------------------------------------------------------
// Weight pre-GEMM:  M[c][f] = sum_o Wc[o][c] * Wf[o][f]   (folds keys-GEMM away)
// ---------------------------------------------------------------------------
__global__ __launch_bounds__(256) void uplatt_make_M(const float* __restrict__ Wc,
                                                     const float* __restrict__ Wf,
                                                     float* __restrict__ Mout) {
  int f = blockIdx.x * 256 + threadIdx.x;   // 0..511
  int c = blockIdx.y;                       // 0..255
  float acc = 0.f;
  for (int o = 0; o < CCH; ++o)
    acc = fmaf(Wc[o * CCH + c], Wf[o * CFH + f], acc);
  Mout[c * CFH + f] = acc;
}

__global__ __launch_bounds__(256) void uplatt_make_bc(const float* __restrict__ Wc,
                                                      const float* __restrict__ Wfb,
                                                      float* __restrict__ bc) {
  int c = threadIdx.x;
  float acc = 0.f;
  for (int o = 0; o < CCH; ++o)
    acc = fmaf(Wc[o * CCH + c], Wfb[o], acc);
  bc[c] = acc;
}

// ---------------------------------------------------------------------------
// Repack Wv [o][f][ky][kx] -> Wr [tap][o][f] so conv A-fetches are contiguous
// b64 loads with 16B stride across K-steps (clause-friendly).
// ---------------------------------------------------------------------------
__global__ __launch_bounds__(256) void uplatt_repack_wv(const float* __restrict__ Wv,
                                                        float* __restrict__ Wr) {
  int idx = blockIdx.x * 256 + threadIdx.x;  // over 9*256*512 = 1,179,648
  int tap = idx >> 17;                       // TAPSZ == 1<<17
  int of  = idx & (TAPSZ - 1);               // o*512 + f
  Wr[idx] = Wv[(size_t)of * 9 + tap];
}

// ---------------------------------------------------------------------------
// qk = M @ mainstream + bc   (per batch: [256,512] x [512,4096])
// One wave computes a 16(c) x 64(p) tile: 4 f32 accumulators, shared A.
// ---------------------------------------------------------------------------
__global__ __launch_bounds__(32) void uplatt_qk_gemm(const float* __restrict__ Mw,
                                                     const float* __restrict__ bc,
                                                     const float* __restrict__ ms,
                                                     float* __restrict__ qk) {
  const int lane = threadIdx.x;
  const int nl = lane & 15;       // N / M row within half
  const int kq = lane >> 4;       // K pair select: {0,1} vs {2,3}
  const int p0 = blockIdx.x * 64;
  const int c0 = blockIdx.y * 16;
  const int b  = blockIdx.z;
  const float* X = ms + (size_t)b * CFH * HWS;
  v8f acc0 = {}, acc1 = {}, acc2 = {}, acc3 = {};
  const float* arow = Mw + (size_t)(c0 + nl) * CFH + kq * 2;
  for (int k = 0; k < CFH; k += 4) {
    v2f a;
    a.x = arow[k];
    a.y = arow[k + 1];
    const float* brow = X + (size_t)(k + kq * 2) * HWS + p0 + nl;
    v2f b0, b1, b2, b3;
    b0.x = brow[0];  b0.y = brow[HWS];
    b1.x = brow[16]; b1.y = brow[HWS + 16];
    b2.x = brow[32]; b2.y = brow[HWS + 32];
    b3.x = brow[48]; b3.y = brow[HWS + 48];
    acc0 = __builtin_amdgcn_wmma_f32_16x16x4_f32(false, a, false, b0, (short)0, acc0, false, false);
    acc1 = __builtin_amdgcn_wmma_f32_16x16x4_f32(false, a, false, b1, (short)0, acc1, false, false);
    acc2 = __builtin_amdgcn_wmma_f32_16x16x4_f32(false, a, false, b2, (short)0, acc2, false, false);
    acc3 = __builtin_amdgcn_wmma_f32_16x16x4_f32(false, a, false, b3, (short)0, acc3, false, false);
  }
  const int mhi = kq * 8;   // D: vgpr r -> M=r (lanes 0-15) / M=r+8 (lanes 16-31)
  #pragma unroll
  for (int r = 0; r < 8; ++r) {
    int c = c0 + r + mhi;
    float bias = bc[c];
    float* orow = qk + (size_t)(b * CCH + c) * HWS + p0 + nl;
    orow[0]  = acc0[r] + bias;
    orow[16] = acc1[r] + bias;
    orow[32] = acc2[r] + bias;
    orow[48] = acc3[r] + bias;
  }
}

// ---------------------------------------------------------------------------
// values = 3x3 conv (SAME) Cf->Cc, implicit GEMM: 9 taps x K=512 WMMA loop.
// One wave computes 16(o) x 32(p); zero-pad via clamp + mask (branch free).
// A comes from repacked Wr[tap][o][f] -> contiguous b64 weight fetches.
// ---------------------------------------------------------------------------
__global__ __launch_bounds__(32) void uplatt_conv3x3(const float* __restrict__ Wr,
                                                     const float* __restrict__ Wvb,
                                                     const float* __restrict__ ms,
                                                     float* __restrict__ vals) {
  const int lane = threadIdx.x;
  const int nl = lane & 15;
  const int kq = lane >> 4;
  const int pb = blockIdx.x;        // 0..127: (row, half-row)
  const int y  = pb >> 1;
  const int xb = (pb & 1) * 32;
  const int o0 = blockIdx.y * 16;
  const int b  = blockIdx.z;
  const float* X = ms + (size_t)b * CFH * HWS;
  v8f acc0 = {}, acc1 = {};
  for (int ky = 0; ky < 3; ++ky) {
    int yy = y + ky - 1;
    bool yok = (unsigned)yy < (unsigned)HSM;
    int yyc = yok ? yy : 0;
    for (int kx = 0; kx < 3; ++kx) {
      int x0i = xb + nl + kx - 1;
      int x1i = x0i + 16;
      float m0 = (yok && (unsigned)x0i < (unsigned)HSM) ? 1.f : 0.f;
      float m1 = (yok && (unsigned)x1i < (unsigned)HSM) ? 1.f : 0.f;
      int x0c = ((unsigned)x0i < (unsigned)HSM) ? x0i : 0;
      int x1c = ((unsigned)x1i < (unsigned)HSM) ? x1i : 0;
      const float* arow = Wr + (size_t)(ky * 3 + kx) * TAPSZ
                             + (size_t)(o0 + nl) * CFH + kq * 2;
      const float* brow = X + ((size_t)(kq * 2) * HSM + yyc) * HSM;
      for (int f = 0; f < CFH; f += 4) {
        v2f a;
        a.x = arow[f];
        a.y = arow[f + 1];
        const float* bp = brow + (size_t)f * HWS;
        v2f b0, b1;
        b0.x = m0 * bp[x0c];
        b0.y = m0 * bp[HWS + x0c];
        b1.x = m1 * bp[x1c];
        b1.y = m1 * bp[HWS + x1c];
        acc0 = __builtin_amdgcn_wmma_f32_16x16x4_f32(false, a, false, b0, (short)0, acc0, false, false);
        acc1 = __builtin_amdgcn_wmma_f32_16x16x4_f32(false, a, false, b1, (short)0, acc1, false, false);
      }
    }
  }
  const int mhi = kq * 8;
  #pragma unroll
  for (int r = 0; r < 8; ++r) {
    int c = o0 + r + mhi;
    float bias = Wvb[c];
    float* orow = vals + (size_t)(b * CCH + c) * HWS + y * HSM + xb + nl;
    orow[0]  = acc0[r] + bias;
    orow[16] = acc1[r] + bias;
  }
}

// ---------------------------------------------------------------------------
// Fused attention: scores (ctx . qk_up) -> softmax over N -> weighted sum + v_up.
// Block = 256 threads handles (b, y, 32 consecutive x). qk tile staged in LDS
// via async-to-LDS copies (ASYNCcnt). Contexts streamed with software prefetch;
// second contexts pass per tile hits L2 (192 MB >> working set).
// ---------------------------------------------------------------------------
__global__ __launch_bounds__(256) void uplatt_attn(const float* __restrict__ ctx,
                                                   const float* __restrict__ qk,
                                                   const float* __restrict__ vals,
                                                   float* __restrict__ out) {
  __shared__ __align__(16) float qk_s[CCH][20];   // 16 half-res x + pad (16B-aligned rows)
  __shared__ float scpart[2][NCTX][32];
  __shared__ float attnw[NCTX][32];
  const int t  = threadIdx.x;
  const int x0 = blockIdx.x * 32;
  const int y  = blockIdx.y;
  const int b  = blockIdx.z;
  const int hy = y >> 1;
  const int hx0 = x0 >> 1;

  // phase 1: async-stage qk[b, :, hy, hx0..hx0+15] into LDS (256 rows x 64B,
  // as 1024 b128 chunks; both sides 16B-aligned).
  for (int idx = t; idx < CCH * 4; idx += 256) {
    int c = idx >> 2, q = idx & 3;
    const float* g = qk + ((size_t)(b * CCH + c) * HSM + hy) * HSM + hx0 + q * 4;
    unsigned loff = (unsigned)(uintptr_t)(&qk_s[c][q * 4]);
    async_b128_to_lds(loff, g);
  }
  wait_asynccnt0();
  __syncthreads();

  // phase 2: partial scores; wave w -> (n = w&3, c-half = w>>2), lane = x.
  // 64KB-stride stream: software-prefetch 16 lines (1MB) ahead.
  {
    int lane = t & 31;
    int w = t >> 5;
    int n = w & 3;
    int ch = w >> 2;
    const float* cp = ctx + ((size_t)((b * NCTX + n) * CCH) + ch * 128) * HWB
                          + (size_t)y * 128 + x0 + lane;
    int hx = lane >> 1;
    float s = 0.f;
    for (int i = 0; i < 128; ++i) {
      if (i + 16 < 128) __builtin_prefetch(cp + (size_t)(i + 16) * HWB, 0, 0);
      s = fmaf(cp[(size_t)i * HWB], qk_s[ch * 128 + i][hx], s);
    }
    scpart[ch][n][lane] = s;
  }
  __syncthreads();

  // phase 3: softmax over the 4 contexts, one thread per x
  if (t < 32) {
    float s0 = scpart[0][0][t] + scpart[1][0][t];
    float s1 = scpart[0][1][t] + scpart[1][1][t];
    float s2 = scpart[0][2][t] + scpart[1][2][t];
    float s3 = scpart[0][3][t] + scpart[1][3][t];
    float m = fmaxf(fmaxf(s0, s1), fmaxf(s2, s3));
    float e0 = __expf(s0 - m), e1 = __expf(s1 - m);
    float e2 = __expf(s2 - m), e3 = __expf(s3 - m);
    float inv = 1.f / (e0 + e1 + e2 + e3);
    attnw[0][t] = e0 * inv;
    attnw[1][t] = e1 * inv;
    attnw[2][t] = e2 * inv;
    attnw[3][t] = e3 * inv;
  }
  __syncthreads();

  // phase 4: out = sum_n attn_n * ctx_n + upsample2x(values)  (L2-hit reread)
  {
    int lane = t & 31;
    int w = t >> 5;
    float a0 = attnw[0][lane], a1 = attnw[1][lane];
    float a2 = attnw[2][lane], a3 = attnw[3][lane];
    int hx = hx0 + (lane >> 1);
    for (int i = 0; i < 32; ++i) {
      int c = w * 32 + i;
      const float* cpix = ctx + ((size_t)(b * NCTX) * CCH + c) * HWB
                              + (size_t)y * 128 + x0 + lane;
      float r = vals[((size_t)(b * CCH + c) * HSM + hy) * HSM + hx];
      r = fmaf(a0, cpix[0], r);
      r = fmaf(a1, cpix[(size_t)CCH * HWB], r);
      r = fmaf(a2, cpix[(size_t)2 * CCH * HWB], r);
      r = fmaf(a3, cpix[(size_t)3 * CCH * HWB], r);
      out[((size_t)(b * CCH + c) * 128 + y) * 128 + x0 + lane] = r;
    }
  }
}

// ---------------------------------------------------------------------------
extern "C" void kernel_launch(void* const* d_in, const int* in_sizes, int n_in,
                              void* d_out, int out_size, void* d_ws, size_t ws_size,
                              hipStream_t stream) {
  const float* contexts   = (const float*)d_in[0];
  const float* mainstream = (const float*)d_in[1];
  const float* Wc_w       = (const float*)d_in[2];
  // d_in[3] = Wc_b: constant-across-n score shift -> cancels in softmax; unused.
  const float* Wf_w       = (const float*)d_in[4];
  const float* Wf_b       = (const float*)d_in[5];
  const float* Wv_w       = (const float*)d_in[6];
  const float* Wv_b       = (const float*)d_in[7];
  float* out = (float*)d_out;

  // workspace: qk 16MB | vals 16MB | M 512KB | bc 1KB(+pad) | Wr 4.5MB  ~ 37 MiB
  char* ws = (char*)d_ws;
  float* qk   = (float*)(ws);
  float* vals = (float*)(ws + (size_t)16777216);
  float* Mw   = (float*)(ws + (size_t)2 * 16777216);
  float* bc   = (float*)(ws + (size_t)2 * 16777216 + 524288);
  float* Wr   = (float*)(ws + (size_t)2 * 16777216 + 524288 + 1024);

  uplatt_make_M   <<<dim3(2, 256, 1), 256, 0, stream>>>(Wc_w, Wf_w, Mw);
  uplatt_make_bc  <<<dim3(1, 1, 1),   256, 0, stream>>>(Wc_w, Wf_b, bc);
  uplatt_repack_wv<<<dim3(4608, 1, 1),256, 0, stream>>>(Wv_w, Wr);
  uplatt_qk_gemm  <<<dim3(64, 16, 4),  32, 0, stream>>>(Mw, bc, mainstream, qk);
  uplatt_conv3x3  <<<dim3(128, 16, 4), 32, 0, stream>>>(Wr, Wv_b, mainstream, vals);
  uplatt_attn     <<<dim3(4, 128, 4), 256, 0, stream>>>(contexts, qk, vals, out);
}